// EdgeModel_86397562127190
// MI455X (gfx1250) — compile-verified
//
#include <hip/hip_runtime.h>

typedef __attribute__((ext_vector_type(2))) float v2f;
typedef __attribute__((ext_vector_type(8))) float v8f;

#define WAVES_PER_BLOCK 8

__global__ __launch_bounds__(256) void edge_mlp_wmma_f32(
    const float* __restrict__ src, const float* __restrict__ dst,
    const float* __restrict__ ea,  const float* __restrict__ u,
    const long long* __restrict__ batch,
    const float* __restrict__ W1, const float* __restrict__ b1,
    const float* __restrict__ W2, const float* __restrict__ b2,
    float* __restrict__ out, int E)
{
    // 1 KB LDS slab per wave for the H (16x16 f32) layout transpose
    __shared__ float lds[WAVES_PER_BLOCK][256];
    const int lane = threadIdx.x & 31;
    const int wv   = threadIdx.x >> 5;
    const int n    = lane & 15;          // column (B/C/D layouts) / row (A layout)
    const bool hi  = lane >= 16;         // second half-wave
    float* hbuf = lds[wv];

    // ---- per-wave constant B tiles (WMMA 32-bit B layout: VGPR0 = K rows {0,2},
    //      VGPR1 = K rows {1,3}; lanes 0-15 -> first row of pair, 16-31 -> second) ----
    const int k0 = hi ? 2 : 0;

    // W1: (4,10) row-major, padded to 4x16
    v2f w1t;
    w1t.x = (n < 10) ? W1[(k0 + 0) * 10 + n] : 0.0f;
    w1t.y = (n < 10) ? W1[(k0 + 1) * 10 + n] : 0.0f;

    // W2: (10,19) row-major, padded to 12x32 as 3 K-chunks x 2 N-tiles
    v2f w2t[3][2];
#pragma unroll
    for (int c = 0; c < 3; ++c) {
#pragma unroll
        for (int t = 0; t < 2; ++t) {
            const int col = t * 16 + n;
            const int ka  = 4 * c + k0;
            const int kb  = ka + 1;
            w2t[c][t].x = (col < 19 && ka < 10) ? W2[ka * 19 + col] : 0.0f;
            w2t[c][t].y = (col < 19 && kb < 10) ? W2[kb * 19 + col] : 0.0f;
        }
    }

    // Biases folded into loop-invariant WMMA accumulator-init registers:
    // C/D element (m, n) lives in lane with column n -> all 8 regs = bias[n].
    const float b1n  = (n < 10) ? b1[n] : 0.0f;
    const float b2n0 = b2[n];
    const float b2n1 = (n < 3) ? b2[16 + n] : 0.0f;
    v8f hinit, o0init, o1init;
#pragma unroll
    for (int r = 0; r < 8; ++r) { hinit[r] = b1n; o0init[r] = b2n0; o1init[r] = b2n1; }

    const int ntiles = E >> 4;
    const int gwave  = blockIdx.x * WAVES_PER_BLOCK + wv;
    const int nwaves = gridDim.x * WAVES_PER_BLOCK;

    // per-lane base pointers; per-row offsets become DS/VMEM immediate offsets
    const int e0 = gwave * 16 + n;
    const float*     ps = src + e0;
    const float*     pd = dst + e0;
    const float*     pe = ea  + e0;
    const long long* pb = batch + e0;
    float*           po = out + (gwave * 16 + (hi ? 8 : 0)) * 19 + n;
    const int  estep = nwaves * 16;
    const long ostep = (long)nwaves * 16 * 19;

    float* stp = &hbuf[(hi ? 128 : 0) + n];  // LDS store base, rows at +64 B
    const float* ldp = &hbuf[n * 16 + k0];   // LDS load base (A layout), chunks at +16 B

    for (int tile = gwave; tile < ntiles; tile += nwaves) {
        // A1 tile: 16 edges x K=4 (A layout: lanes 0-15 hold K=0,1; 16-31 hold K=2,3)
        v2f a1;
        if (!hi) { a1.x = ps[0]; a1.y = pd[0]; }
        else     { a1.x = pe[0]; a1.y = u[(int)pb[0]]; }

        // ---- layer 1: H = x @ W1 + b1 (bias in accumulator init) ----
        v8f h = __builtin_amdgcn_wmma_f32_16x16x4_f32(false, a1, false, w1t,
                                                      (short)0, hinit, false, false);
        // ReLU, spill H (C/D layout) to LDS row-major; row offsets are immediates
#pragma unroll
        for (int r = 0; r < 8; ++r)
            stp[r * 16] = fmaxf(h[r], 0.0f);
        asm volatile("s_wait_dscnt 0x0" ::: "memory");  // cross-lane LDS RAW

        // ---- layer 2: O = H @ W2 + b2 (K=12; H cols 10..15 are zero) ----
        v2f a2_0 = *(const v2f*)(ldp + 0);
        v2f a2_1 = *(const v2f*)(ldp + 4);
        v2f a2_2 = *(const v2f*)(ldp + 8);
        v8f o0 = __builtin_amdgcn_wmma_f32_16x16x4_f32(false, a2_0, false, w2t[0][0],
                                                       (short)0, o0init, false, false);
        v8f o1 = __builtin_amdgcn_wmma_f32_16x16x4_f32(false, a2_0, false, w2t[0][1],
                                                       (short)0, o1init, false, false);
        o0 = __builtin_amdgcn_wmma_f32_16x16x4_f32(false, a2_1, false, w2t[1][0],
                                                   (short)0, o0, false, false);
        o1 = __builtin_amdgcn_wmma_f32_16x16x4_f32(false, a2_1, false, w2t[1][1],
                                                   (short)0, o1, false, false);
        o0 = __builtin_amdgcn_wmma_f32_16x16x4_f32(false, a2_2, false, w2t[2][0],
                                                   (short)0, o0, false, false);
        o1 = __builtin_amdgcn_wmma_f32_16x16x4_f32(false, a2_2, false, w2t[2][1],
                                                   (short)0, o1, false, false);

        // store: cols 0..15 for all lanes, cols 16..18 for n<3; row offsets immediates
#pragma unroll
        for (int r = 0; r < 8; ++r)
            po[r * 19] = o0[r];
        if (n < 3) {
#pragma unroll
            for (int r = 0; r < 8; ++r)
                po[r * 19 + 16] = o1[r];
        }

        ps += estep; pd += estep; pe += estep; pb += estep; po += ostep;
    }

    // scalar tail (dead for E = 8M, kept for robustness)
    const int rem = E & 15;
    if (rem && gwave == 0 && lane < rem) {
        const int e = (ntiles << 4) + lane;
        const float x0 = src[e], x1 = dst[e], x2 = ea[e], x3 = u[(int)batch[e]];
        float hh[10];
#pragma unroll
        for (int j = 0; j < 10; ++j) {
            float v = x0 * W1[j] + x1 * W1[10 + j] + x2 * W1[20 + j] + x3 * W1[30 + j] + b1[j];
            hh[j] = v > 0.0f ? v : 0.0f;
        }
        for (int q = 0; q < 19; ++q) {
            float v = b2[q];
#pragma unroll
            for (int j = 0; j < 10; ++j) v += hh[j] * W2[j * 19 + q];
            out[e * 19 + q] = v;
        }
    }
}

extern "C" void kernel_launch(void* const* d_in, const int* in_sizes, int n_in,
                              void* d_out, int out_size, void* d_ws, size_t ws_size,
                              hipStream_t stream) {
    const float*     src   = (const float*)d_in[0];
    const float*     dst   = (const float*)d_in[1];
    const float*     ea    = (const float*)d_in[2];
    const float*     u     = (const float*)d_in[3];
    const long long* batch = (const long long*)d_in[4];  // int64 per reference
    const float*     W1    = (const float*)d_in[5];
    const float*     b1    = (const float*)d_in[6];
    const float*     W2    = (const float*)d_in[7];
    const float*     b2    = (const float*)d_in[8];
    float* out = (float*)d_out;

    const int E = in_sizes[0];
    const int ntiles = E / 16;
    int blocks = (ntiles + WAVES_PER_BLOCK - 1) / WAVES_PER_BLOCK;
    if (blocks > 4096) blocks = 4096;   // ~32K waves, grid-stride over tiles
    if (blocks < 1)    blocks = 1;

    edge_mlp_wmma_f32<<<blocks, 256, 0, stream>>>(src, dst, ea, u, batch,
                                                  W1, b1, W2, b2, out, E);
}